// GraphGCN_82463372083418
// MI455X (gfx1250) — compile-verified
//
#include <hip/hip_runtime.h>
#include <hip/hip_bf16.h>
#include <math.h>

typedef __attribute__((ext_vector_type(2))) float v2f;
typedef __attribute__((ext_vector_type(8))) float v8f;

// ---------------------------------------------------------------------------
// 1) degree init (self-loop contributes 1 to every node)
__global__ void deg_init_kernel(float* __restrict__ deg, int n) {
    int i = blockIdx.x * blockDim.x + threadIdx.x;
    if (i < n) deg[i] = 1.0f;
}

// 2) degree count over destination (col) of real edges
__global__ void deg_count_kernel(const long long* __restrict__ ei, float* __restrict__ deg,
                                 int E) {
    int e = blockIdx.x * blockDim.x + threadIdx.x;
    if (e >= E) return;
    int c = (int)ei[(size_t)E + e];
    atomicAdd(&deg[c], 1.0f);
}

// 3) dinv = deg^-0.5 (deg >= 1 always, thanks to self-loops)
__global__ void dinv_kernel(float* __restrict__ dinv, int n) {
    int i = blockIdx.x * blockDim.x + threadIdx.x;
    if (i < n) dinv[i] = rsqrtf(dinv[i]);
}

// ---------------------------------------------------------------------------
// 4) h = x @ W1 via V_WMMA_F32_16X16X4_F32.
//    One wave32 computes one 16x16 output tile (16 nodes x 16 features),
//    marching K=128 in steps of 4 -> 32 WMMA ops per wave.
//    A 16x4 f32 layout: lanes 0-15 -> M=lane, {v0,v1}={K0,K1};
//                       lanes 16-31 -> M=lane-16, {v0,v1}={K2,K3}.
//    B 4x16 f32 layout: lanes 0-15 -> N=lane, {v0,v1}={K0,K1};
//                       lanes 16-31 -> N=lane-16, {v0,v1}={K2,K3}.
//    C/D 16x16 f32: lane<16 -> N=lane, vj = M=j; lane>=16 -> N=lane-16, vj = M=j+8.
__global__ void gemm_xw1_kernel(const float* __restrict__ x, const float* __restrict__ W1,
                                float* __restrict__ h, int ntiles) {
    __shared__ float sW[128 * 16];
    for (int i = threadIdx.x; i < 128 * 16; i += blockDim.x) sW[i] = W1[i];
    __syncthreads();

    int wave = threadIdx.x >> 5;
    int lane = threadIdx.x & 31;
    int tile = blockIdx.x * (blockDim.x >> 5) + wave;
    if (tile >= ntiles) return;  // wave-uniform: EXEC stays all-ones inside

    int m0 = tile << 4;
    int r  = lane & 15;   // M (for A) / N (for B,D)
    int kh = lane >> 4;   // which K pair within the 4-wide step

    const float* xrow = x + (size_t)(m0 + r) * 128;
    v8f c = {};
#pragma unroll
    for (int k0 = 0; k0 < 128; k0 += 4) {
        int k = k0 + (kh << 1);
        v2f a = *(const v2f*)(xrow + k);            // x[m0+r][k], x[m0+r][k+1]
        v2f b;
        b.x = sW[k * 16 + r];                       // W1[k][r]
        b.y = sW[(k + 1) * 16 + r];                 // W1[k+1][r]
        c = __builtin_amdgcn_wmma_f32_16x16x4_f32(
                /*neg_a=*/false, a, /*neg_b=*/false, b,
                /*c_mod=*/(short)0, c, /*reuse_a=*/false, /*reuse_b=*/false);
    }

    int mbase = m0 + (kh << 3);
#pragma unroll
    for (int j = 0; j < 8; ++j)
        h[(size_t)(mbase + j) * 16 + r] = c[j];
}

// ---------------------------------------------------------------------------
// 5) layer-1 aggregation init with self-loop term: agg[i][f] = h[i][f]*dinv[i]^2
__global__ void self1_kernel(const float* __restrict__ h, const float* __restrict__ dinv,
                             float* __restrict__ agg, int n) {
    int idx = blockIdx.x * blockDim.x + threadIdx.x;
    if (idx >= n * 16) return;
    int i = idx >> 4;
    float d = dinv[i];
    agg[idx] = h[idx] * d * d;
}

// 6) layer-1 edge scatter: agg[col][f] += h[row][f] * dinv[row]*dinv[col]
//    one thread per (edge, feature); a wave32 covers 2 edges x 16 features.
__global__ void edges1_kernel(const long long* __restrict__ ei, const float* __restrict__ h,
                              const float* __restrict__ dinv, float* __restrict__ agg,
                              int E) {
    long long idx = (long long)blockIdx.x * blockDim.x + threadIdx.x;
    int e = (int)(idx >> 4);
    if (e >= E) return;
    int f = (int)(idx & 15);
    int r = (int)ei[e];
    int c = (int)ei[(size_t)E + e];
    float norm = dinv[r] * dinv[c];
    atomicAdd(&agg[(size_t)c * 16 + f], h[(size_t)r * 16 + f] * norm);
}

// 7) per node: relu(agg+b1) . W2  -> z ; init out accumulator with self-loop
__global__ void node2_kernel(const float* __restrict__ agg, const float* __restrict__ b1,
                             const float* __restrict__ W2, const float* __restrict__ dinv,
                             float* __restrict__ z, float* __restrict__ outAcc, int n) {
    int i = blockIdx.x * blockDim.x + threadIdx.x;
    if (i >= n) return;
    float acc = 0.0f;
#pragma unroll
    for (int f = 0; f < 16; ++f) {
        float v = agg[(size_t)i * 16 + f] + b1[f];
        v = fmaxf(v, 0.0f);              // relu (dropout is identity in eval)
        acc = fmaf(v, W2[f], acc);
    }
    z[i] = acc;
    float d = dinv[i];
    outAcc[i] = acc * d * d;             // self-loop contribution of layer 2
}

// 8) layer-2 edge scatter: outAcc[col] += z[row] * dinv[row]*dinv[col]
__global__ void edges2_kernel(const long long* __restrict__ ei, const float* __restrict__ z,
                              const float* __restrict__ dinv, float* __restrict__ outAcc,
                              int E) {
    int e = blockIdx.x * blockDim.x + threadIdx.x;
    if (e >= E) return;
    int r = (int)ei[e];
    int c = (int)ei[(size_t)E + e];
    atomicAdd(&outAcc[c], z[r] * dinv[r] * dinv[c]);
}

// 9) out = sigmoid(outAcc + b2)
__global__ void sigmoid_kernel(const float* __restrict__ outAcc, const float* __restrict__ b2,
                               float* __restrict__ out, int n) {
    int i = blockIdx.x * blockDim.x + threadIdx.x;
    if (i >= n) return;
    float v = outAcc[i] + b2[0];
    out[i] = 1.0f / (1.0f + __expf(-v));
}

// ---------------------------------------------------------------------------
extern "C" void kernel_launch(void* const* d_in, const int* in_sizes, int n_in,
                              void* d_out, int out_size, void* d_ws, size_t ws_size,
                              hipStream_t stream) {
    const float*     x  = (const float*)d_in[0];
    const long long* ei = (const long long*)d_in[1];
    const float*     W1 = (const float*)d_in[2];
    const float*     b1 = (const float*)d_in[3];
    const float*     W2 = (const float*)d_in[4];
    const float*     b2 = (const float*)d_in[5];

    const int n = in_sizes[0] / 128;   // 100000
    const int E = in_sizes[1] / 2;     // 3200000

    // workspace layout (floats): dinv[n] | h[16n] | agg[16n] | z[n] | outAcc[n]
    float* base   = (float*)d_ws;
    float* dinv   = base;
    float* h      = dinv + n;
    float* agg    = h + (size_t)16 * n;
    float* z      = agg + (size_t)16 * n;
    float* outAcc = z + n;
    float* out    = (float*)d_out;

    const int B = 256;
    const int gN   = (n + B - 1) / B;
    const int gNF  = (n * 16 + B - 1) / B;
    const int gE   = (E + B - 1) / B;
    const int gEF  = (int)(((long long)E * 16 + B - 1) / B);

    const int ntiles = n / 16;                         // n is a multiple of 16
    const int wavesPerBlock = B / 32;
    const int gGemm = (ntiles + wavesPerBlock - 1) / wavesPerBlock;

    deg_init_kernel <<<gN,   B, 0, stream>>>(dinv, n);
    deg_count_kernel<<<gE,   B, 0, stream>>>(ei, dinv, E);
    dinv_kernel     <<<gN,   B, 0, stream>>>(dinv, n);
    gemm_xw1_kernel <<<gGemm,B, 0, stream>>>(x, W1, h, ntiles);
    self1_kernel    <<<gNF,  B, 0, stream>>>(h, dinv, agg, n);
    edges1_kernel   <<<gEF,  B, 0, stream>>>(ei, h, dinv, agg, E);
    node2_kernel    <<<gN,   B, 0, stream>>>(agg, b1, W2, dinv, z, outAcc, n);
    edges2_kernel   <<<gE,   B, 0, stream>>>(ei, z, dinv, outAcc, E);
    sigmoid_kernel  <<<gN,   B, 0, stream>>>(outAcc, b2, out, n);
}